// SparseAttention_77764677861674
// MI455X (gfx1250) — compile-verified
//
#include <hip/hip_runtime.h>
#include <hip/hip_bf16.h>

typedef __attribute__((ext_vector_type(2))) float v2f;
typedef __attribute__((ext_vector_type(8))) float v8f;

#define SEQ_LEN   2048
#define DMODEL    1024
#define NHEADS    16
#define DHEAD     64

// ---------------------------------------------------------------------------
// Generic WMMA f32 GEMM:  C = alpha * (A . B^T) + bias
//   A: M x K, row-major, lda       (K contiguous)
//   B: N x K, row-major, ldb       (K contiguous)  -> C[m,n] = sum_k A[m,k]*B[n,k]
// Block tile: 128(M) x 64(N), K step 16, double-buffered LDS (one barrier per
// K-step). 256 threads = 8 wave32 (4x2 waves); each wave computes a 32x32 tile
// via 4 accumulators of v_wmma_f32_16x16x4_f32.
// store modes: 0 = C[m*ldc+n]
//              1 = head-split      C[(n>>6)*SEQ*64 + m*64 + (n&63)]   (q,k)
//              2 = head-split-T    C[(n>>6)*64*SEQ + (n&63)*SEQ + m]  (v^T)
// ---------------------------------------------------------------------------
#define LDSA_STRIDE 20
#define LDSB_STRIDE 20
#define LDSA_SZ     (128 * LDSA_STRIDE)
#define LDSB_SZ     (64  * LDSB_STRIDE)

__global__ __launch_bounds__(256) void gemm_wmma_f32_kernel(
    const float* __restrict__ A, const float* __restrict__ B,
    const float* __restrict__ bias, float* __restrict__ C,
    int K, int lda, int ldb, int ldc,
    long long sA, long long sB, long long sC,
    float alpha, int mode)
{
    __shared__ float lds_a[2 * LDSA_SZ];
    __shared__ float lds_b[2 * LDSB_SZ];

    const int t    = threadIdx.x;
    const int lane = t & 31;
    const int wave = t >> 5;
    const int wm   = wave & 3;   // 4 waves along M
    const int wn   = wave >> 2;  // 2 waves along N
    const int lh   = lane >> 4;  // half-wave (0/1)
    const int ll   = lane & 15;

    const int z  = blockIdx.z;
    A += (long long)z * sA;
    B += (long long)z * sB;
    C += (long long)z * sC;

    const int m0 = blockIdx.x * 128;
    const int n0 = blockIdx.y * 64;

    // per-thread staging slots (global -> regs -> LDS)
    const int a_row0 = t >> 2;            // rows 0..63   (i = t)
    const int a_row1 = (t + 256) >> 2;    // rows 64..127 (i = t + 256)
    const int a_c4   = (t & 3) << 2;
    const int b_row  = t >> 2;
    const int b_c4   = (t & 3) << 2;

    v8f acc[2][2];
    #pragma unroll
    for (int i = 0; i < 2; ++i)
        #pragma unroll
        for (int j = 0; j < 2; ++j)
            acc[i][j] = (v8f){0.f,0.f,0.f,0.f,0.f,0.f,0.f,0.f};

    const int nsteps = K >> 4;

    // ---- prologue: stage tile 0 into buffer 0
    {
        const float4 va0 = *(const float4*)(A + (long long)(m0 + a_row0) * lda + a_c4);
        const float4 va1 = *(const float4*)(A + (long long)(m0 + a_row1) * lda + a_c4);
        const float4 vb  = *(const float4*)(B + (long long)(n0 + b_row)  * ldb + b_c4);
        *(float4*)(&lds_a[a_row0 * LDSA_STRIDE + a_c4]) = va0;
        *(float4*)(&lds_a[a_row1 * LDSA_STRIDE + a_c4]) = va1;
        *(float4*)(&lds_b[b_row  * LDSB_STRIDE + b_c4]) = vb;
    }

    for (int i = 0; i < nsteps; ++i) {
        __syncthreads();   // tile i visible in buffer (i&1); prior reads of the
                           // buffer we are about to overwrite have completed

        // ---- issue global loads for tile i+1 (overlaps with WMMA below)
        float4 ra0, ra1, rb;
        const bool more = (i + 1) < nsteps;
        if (more) {
            const int k0n = (i + 1) << 4;
            ra0 = *(const float4*)(A + (long long)(m0 + a_row0) * lda + k0n + a_c4);
            ra1 = *(const float4*)(A + (long long)(m0 + a_row1) * lda + k0n + a_c4);
            rb  = *(const float4*)(B + (long long)(n0 + b_row)  * ldb + k0n + b_c4);
        }

        // ---- compute on buffer (i&1)
        const float* __restrict__ la = &lds_a[(i & 1) * LDSA_SZ];
        const float* __restrict__ lb = &lds_b[(i & 1) * LDSB_SZ];
        #pragma unroll
        for (int j = 0; j < 4; ++j) {
            const int koff = j * 4 + lh * 2;
            v2f a0 = *(const v2f*)&la[(wm * 32 +  0 + ll) * LDSA_STRIDE + koff];
            v2f a1 = *(const v2f*)&la[(wm * 32 + 16 + ll) * LDSA_STRIDE + koff];
            v2f b0 = *(const v2f*)&lb[(wn * 32 +  0 + ll) * LDSB_STRIDE + koff];
            v2f b1 = *(const v2f*)&lb[(wn * 32 + 16 + ll) * LDSB_STRIDE + koff];
            acc[0][0] = __builtin_amdgcn_wmma_f32_16x16x4_f32(false, a0, false, b0, (short)0, acc[0][0], false, false);
            acc[0][1] = __builtin_amdgcn_wmma_f32_16x16x4_f32(false, a0, false, b1, (short)0, acc[0][1], false, false);
            acc[1][0] = __builtin_amdgcn_wmma_f32_16x16x4_f32(false, a1, false, b0, (short)0, acc[1][0], false, false);
            acc[1][1] = __builtin_amdgcn_wmma_f32_16x16x4_f32(false, a1, false, b1, (short)0, acc[1][1], false, false);
        }

        // ---- stage tile i+1 into the other buffer (disjoint from compute buffer;
        //      cross-wave safety provided by the top-of-loop barrier)
        if (more) {
            float* __restrict__ na = &lds_a[((i + 1) & 1) * LDSA_SZ];
            float* __restrict__ nb = &lds_b[((i + 1) & 1) * LDSB_SZ];
            *(float4*)(&na[a_row0 * LDSA_STRIDE + a_c4]) = ra0;
            *(float4*)(&na[a_row1 * LDSA_STRIDE + a_c4]) = ra1;
            *(float4*)(&nb[b_row  * LDSB_STRIDE + b_c4]) = rb;
        }
    }

    // ---- store: C/D layout: VGPR r, lane l -> M = r + (l>=16)*8, N = l&15
    #pragma unroll
    for (int mt = 0; mt < 2; ++mt) {
        #pragma unroll
        for (int nt = 0; nt < 2; ++nt) {
            const int n  = n0 + wn * 32 + nt * 16 + ll;
            const float bv = bias ? bias[n] : 0.0f;
            #pragma unroll
            for (int r = 0; r < 8; ++r) {
                const int m = m0 + wm * 32 + mt * 16 + lh * 8 + r;
                const float v = acc[mt][nt][r] * alpha + bv;
                long long idx;
                if (mode == 0)      idx = (long long)m * ldc + n;
                else if (mode == 1) idx = (long long)(n >> 6) * (SEQ_LEN * DHEAD) + (long long)m * DHEAD + (n & 63);
                else                idx = (long long)(n >> 6) * (DHEAD * SEQ_LEN) + (long long)(n & 63) * SEQ_LEN + m;
                C[idx] = v;
            }
        }
    }
}

// ---------------------------------------------------------------------------
// Per-row exact top-k threshold + softmax, in place on attn (16*2048 rows of 2048).
// MSB-first 8-bit radix select on sign-flipped float keys, then
// attn = (s >= kth ? exp(s - max) : 0) / sum   (== reference: NEG_INF mask + softmax)
// ---------------------------------------------------------------------------
__global__ __launch_bounds__(256) void topk_softmax_kernel(
    float* __restrict__ attn, const int* __restrict__ ktop_p)
{
    const int S = SEQ_LEN;
    __shared__ unsigned keys[SEQ_LEN];
    __shared__ unsigned hist[256];
    __shared__ float    red[256];
    __shared__ unsigned s_pref, s_remk;

    const int t = threadIdx.x;
    float* row = attn + (long long)blockIdx.x * S;

    int kv = *ktop_p;
    if (kv > S) kv = S;
    if (kv < 1) kv = 1;

    float vals[8];
    float mx = -3.4e38f;
    #pragma unroll
    for (int i = 0; i < 8; ++i) {
        const float f = row[t + i * 256];
        vals[i] = f;
        mx = fmaxf(mx, f);
        const unsigned u = __float_as_uint(f);
        keys[t + i * 256] = (u & 0x80000000u) ? ~u : (u | 0x80000000u);
    }
    red[t] = mx;
    __syncthreads();
    for (int s = 128; s > 0; s >>= 1) {
        if (t < s) red[t] = fmaxf(red[t], red[t + s]);
        __syncthreads();
    }
    const float rmax = red[0];
    __syncthreads();

    unsigned prefix = 0u;
    unsigned remk   = (unsigned)kv;
    #pragma unroll
    for (int pass = 0; pass < 4; ++pass) {
        const int shift = 24 - 8 * pass;
        hist[t] = 0u;
        __syncthreads();
        #pragma unroll
        for (int i = 0; i < 8; ++i) {
            const unsigned key = keys[t + i * 256];
            const unsigned hi  = (pass == 0) ? 0u : (key >> (shift + 8));
            if (hi == prefix)
                atomicAdd(&hist[(key >> shift) & 255u], 1u);
        }
        __syncthreads();
        if (t == 0) {
            unsigned acc = 0u;
            int b = 255;
            for (; b > 0; --b) {
                const unsigned h = hist[b];
                if (acc + h >= remk) break;
                acc += h;
            }
            s_pref = (prefix << 8) | (unsigned)b;
            s_remk = remk - acc;
        }
        __syncthreads();
        prefix = s_pref;
        remk   = s_remk;
        __syncthreads();
    }

    const unsigned tk = prefix;                 // key of the k-th largest value
    const unsigned uu = (tk & 0x80000000u) ? (tk & 0x7fffffffu) : ~tk;
    const float thresh = __uint_as_float(uu);

    float e[8];
    float lsum = 0.0f;
    #pragma unroll
    for (int i = 0; i < 8; ++i) {
        const float f  = vals[i];
        const float ev = (f >= thresh) ? __expf(f - rmax) : 0.0f;
        e[i] = ev;
        lsum += ev;
    }
    red[t] = lsum;
    __syncthreads();
    for (int s = 128; s > 0; s >>= 1) {
        if (t < s) red[t] += red[t + s];
        __syncthreads();
    }
    const float inv = 1.0f / red[0];
    #pragma unroll
    for (int i = 0; i < 8; ++i)
        row[t + i * 256] = e[i] * inv;
}

// ---------------------------------------------------------------------------
extern "C" void kernel_launch(void* const* d_in, const int* in_sizes, int n_in,
                              void* d_out, int out_size, void* d_ws, size_t ws_size,
                              hipStream_t stream)
{
    const float* Q   = (const float*)d_in[0];
    const float* Kin = (const float*)d_in[1];
    const float* V   = (const float*)d_in[2];
    const float* Wq  = (const float*)d_in[3];
    const float* bq  = (const float*)d_in[4];
    const float* Wk  = (const float*)d_in[5];
    const float* bk  = (const float*)d_in[6];
    const float* Wv  = (const float*)d_in[7];
    const float* bv  = (const float*)d_in[8];
    const float* Wo  = (const float*)d_in[9];
    const float* bo  = (const float*)d_in[10];
    const int*   kt  = (const int*)d_in[11];

    float* out  = (float*)d_out;                               // (2048, 1024)
    float* attn = out + (long long)SEQ_LEN * DMODEL;           // (16, 2048, 2048)

    float* ws   = (float*)d_ws;
    const long long HSZ = (long long)NHEADS * SEQ_LEN * DHEAD; // 2M floats per tensor
    float* q_h  = ws;              // [h][s][d]
    float* k_h  = q_h + HSZ;       // [h][s][d]
    float* vT_h = k_h + HSZ;       // [h][d][s]
    float* ctx  = vT_h + HSZ;      // [s][h*64+d] = (2048, 1024)

    const dim3 blk(256);
    const float scale = 0.125f;    // 1/sqrt(64)

    // 1-3) projections: X (2048x1024) . W^T (1024x1024) + b, head-split stores
    gemm_wmma_f32_kernel<<<dim3(SEQ_LEN/128, DMODEL/64, 1), blk, 0, stream>>>(
        Q,   Wq, bq, q_h,  DMODEL, DMODEL, DMODEL, 0, 0, 0, 0, 1.0f, 1);
    gemm_wmma_f32_kernel<<<dim3(SEQ_LEN/128, DMODEL/64, 1), blk, 0, stream>>>(
        Kin, Wk, bk, k_h,  DMODEL, DMODEL, DMODEL, 0, 0, 0, 0, 1.0f, 1);
    gemm_wmma_f32_kernel<<<dim3(SEQ_LEN/128, DMODEL/64, 1), blk, 0, stream>>>(
        V,   Wv, bv, vT_h, DMODEL, DMODEL, DMODEL, 0, 0, 0, 0, 1.0f, 2);

    // 4) scores = 0.125 * q_h . k_h^T per head, written straight into d_out's attn slab
    gemm_wmma_f32_kernel<<<dim3(SEQ_LEN/128, SEQ_LEN/64, NHEADS), blk, 0, stream>>>(
        q_h, k_h, nullptr, attn, DHEAD, DHEAD, DHEAD, SEQ_LEN,
        (long long)SEQ_LEN * DHEAD, (long long)SEQ_LEN * DHEAD,
        (long long)SEQ_LEN * SEQ_LEN, scale, 0);

    // 5) exact per-row top-k threshold + softmax, in place
    topk_softmax_kernel<<<dim3(NHEADS * SEQ_LEN), blk, 0, stream>>>(attn, kt);

    // 6) ctx = attn . v per head  (B = v^T, row-major over s) -> ctx (2048, 1024)
    gemm_wmma_f32_kernel<<<dim3(SEQ_LEN/128, DHEAD/64, NHEADS), blk, 0, stream>>>(
        attn, vT_h, nullptr, ctx, SEQ_LEN, SEQ_LEN, SEQ_LEN, DMODEL,
        (long long)SEQ_LEN * SEQ_LEN, (long long)DHEAD * SEQ_LEN,
        (long long)DHEAD, 1.0f, 0);

    // 7) out = ctx . Wo^T + bo
    gemm_wmma_f32_kernel<<<dim3(SEQ_LEN/128, DMODEL/64, 1), blk, 0, stream>>>(
        ctx, Wo, bo, out, DMODEL, DMODEL, DMODEL, DMODEL, 0, 0, 0, 1.0f, 0);
}